// LSTMNet_3728031613714
// MI455X (gfx1250) — compile-verified
//
#include <hip/hip_runtime.h>
#include <hip/hip_bf16.h>

// Problem constants (from reference)
#define Dh   100          // hidden size
#define Bb   256          // batch
#define Tt   512          // seq len
#define Gg   400          // 4*Dh gates
#define NT   25           // Gg/16 N-tiles
#define ROWS 16           // batch rows per scan workgroup

typedef float v2f __attribute__((ext_vector_type(2)));
typedef float v8f __attribute__((ext_vector_type(8)));

// f32 WMMA 16x16x4: D = A(16x4) * B(4x16) + C(16x16), full f32 precision.
// 8-arg pattern: (neg_a, A, neg_b, B, c_mod, C, reuse_a, reuse_b)
__device__ __forceinline__ v8f wmma_f32(v2f a, v2f b, v8f c) {
    return __builtin_amdgcn_wmma_f32_16x16x4_f32(
        false, a, false, b, (short)0, c, false, false);
}

__device__ __forceinline__ float sigmoid_f(float x) {
    return 1.0f / (1.0f + __expf(-x));
}
__device__ __forceinline__ float tanh_f(float x) {
    // 1 - 2/(e^{2x}+1): well-behaved at +/-inf of exp
    float e = __expf(2.0f * x);
    return 1.0f - 2.0f / (e + 1.0f);
}

// ---------------------------------------------------------------------------
// Phase 1: xg[m, g] = emb[x[m]] @ w_ih^T + (b_ih + b_hh),  m = b*T + t
// One wave per 16(M) x 16(N) tile, K=100 in 25 WMMA steps of K=4.
// Block = 256 threads = 8 waves = 8 consecutive M-tiles sharing one N-tile
// (B-fragment loads identical across waves -> served from cache).
// ---------------------------------------------------------------------------
__global__ void lstm_input_gemm(const int* __restrict__ x,
                                const float* __restrict__ emb,
                                const float* __restrict__ w_ih,
                                const float* __restrict__ b_ih,
                                const float* __restrict__ b_hh,
                                float* __restrict__ xg) {
    const int lane = threadIdx.x & 31;
    const int wave = threadIdx.x >> 5;
    const int ntile = blockIdx.x % NT;
    const int mgrp  = blockIdx.x / NT;
    const int m0 = (mgrp * 8 + wave) * 16;
    const int n0 = ntile * 16;
    const int row = lane & 15;   // A: M row / B: N col for this lane
    const int hi  = lane >> 4;   // lane half selects K pair
    const int koff = 2 * hi;

    const int   token = x[m0 + row];
    const float* arow = emb  + (size_t)token * Dh;        // A[m][k] = xe[m][k]
    const float* brow = w_ih + (size_t)(n0 + row) * Dh;   // B[k][n] = w_ih[n][k]

    v8f acc = {};
#pragma unroll
    for (int kb = 0; kb < Dh; kb += 4) {
        v2f a = *(const v2f*)(arow + kb + koff);   // aligned global_load_b64
        v2f b = *(const v2f*)(brow + kb + koff);
        acc = wmma_f32(a, b, acc);
    }

    const float bias = b_ih[n0 + row] + b_hh[n0 + row];
#pragma unroll
    for (int r = 0; r < 8; ++r) {                  // D: M = r + 8*hi, N = row
        xg[(size_t)(m0 + r + 8 * hi) * Gg + (n0 + row)] = acc[r] + bias;
    }
}

// ---------------------------------------------------------------------------
// Phase 2: sequential LSTM scan. Batch rows are independent recurrences ->
// 16 workgroups x 16 batch rows, zero cross-WG sync. w_hh (160 KB) + h/c/
// hsum/gates all LDS-resident (~200 KB of the 320 KB WGP LDS).
// Per step: gates = xg[:,t,:] (as WMMA C-matrix) + h @ w_hh^T via WMMA,
// then elementwise cell update; 2 workgroup barriers per step.
// ---------------------------------------------------------------------------
__global__ void lstm_scan(const float* __restrict__ xg,
                          const float* __restrict__ w_hh,
                          const float* __restrict__ fc_w,
                          const float* __restrict__ fc_b,
                          const float* __restrict__ h0,
                          const float* __restrict__ c0,
                          float* __restrict__ out) {
    extern __shared__ float smem[];
    float* whh_s  = smem;                    // 40000 f (160 KB)
    float* h_s    = whh_s + Gg * Dh;         // 1600 f
    float* c_s    = h_s   + ROWS * Dh;       // 1600 f
    float* hsum_s = c_s   + ROWS * Dh;       // 1600 f
    float* gate_s = hsum_s + ROWS * Dh;      // 6400 f

    const int tid  = threadIdx.x;
    const int lane = tid & 31;
    const int wave = tid >> 5;
    const int b0   = blockIdx.x * ROWS;

    for (int i = tid; i < Gg * Dh; i += 256) whh_s[i] = w_hh[i];
    for (int i = tid; i < ROWS * Dh; i += 256) {
        int r = i / Dh, d = i % Dh;
        h_s[i]    = h0[(size_t)(b0 + r) * Dh + d];
        c_s[i]    = c0[(size_t)(b0 + r) * Dh + d];
        hsum_s[i] = 0.0f;
    }
    __syncthreads();

    const int row  = lane & 15;
    const int hi   = lane >> 4;
    const int koff = 2 * hi;

    for (int t = 0; t < Tt; ++t) {
        // prefetch next timestep's 16x400 xg tile (global_prefetch_b8)
        if (t + 1 < Tt) {
            int pr = tid >> 4;                // 0..15 rows
            int pc = (tid & 15) * 25;         // 100B spacing covers 1600B row
            __builtin_prefetch(
                xg + ((size_t)(b0 + pr) * Tt + (t + 1)) * Gg + pc, 0, 0);
        }

        // --- gates = xg_t + h @ w_hh^T  (WMMA, 25 N-tiles over 8 waves) ---
        for (int nt = wave; nt < NT; nt += 8) {
            const int n0 = nt * 16;
            v8f acc;
#pragma unroll
            for (int r = 0; r < 8; ++r)       // C init = xg tile (has biases)
                acc[r] = xg[((size_t)(b0 + r + 8 * hi) * Tt + t) * Gg
                            + n0 + row];
            const float* ha = h_s   + row * Dh + koff;         // A[m][k]=h
            const float* bb = whh_s + (n0 + row) * Dh + koff;  // B[k][n]=whh[n][k]
#pragma unroll
            for (int kb = 0; kb < Dh; kb += 4) {
                v2f a = *(const v2f*)(ha + kb);   // ds_read_b64
                v2f b = *(const v2f*)(bb + kb);
                acc = wmma_f32(a, b, acc);
            }
#pragma unroll
            for (int r = 0; r < 8; ++r)
                gate_s[(r + 8 * hi) * Gg + n0 + row] = acc[r];
        }
        __syncthreads();

        // --- cell update: i,f,g,o chunks of Dh along the gate axis ---
        for (int i = tid; i < ROWS * Dh; i += 256) {
            int r = i / Dh, d = i % Dh;
            const float* gr = gate_s + r * Gg + d;
            float ig = sigmoid_f(gr[0]);
            float fg = sigmoid_f(gr[Dh]);
            float gg = tanh_f(gr[2 * Dh]);
            float og = sigmoid_f(gr[3 * Dh]);
            float c  = fmaf(fg, c_s[i], ig * gg);
            c_s[i]   = c;
            float h  = og * tanh_f(c);
            h_s[i]   = h;
            hsum_s[i] += h;
        }
        __syncthreads();
    }

    // --- head: out = (hsum/T) @ fc_w^T + fc_b   (16 rows x 2 outputs) ---
    if (tid < 32) {
        int r = tid >> 1, j = tid & 1;
        float s = 0.0f;
        for (int d = 0; d < Dh; ++d)
            s = fmaf(hsum_s[r * Dh + d], fc_w[j * Dh + d], s);
        out[(b0 + r) * 2 + j] = s * (1.0f / Tt) + fc_b[j];
    }
}

// ---------------------------------------------------------------------------
extern "C" void kernel_launch(void* const* d_in, const int* in_sizes, int n_in,
                              void* d_out, int out_size, void* d_ws, size_t ws_size,
                              hipStream_t stream) {
    const int*   x    = (const int*)d_in[0];
    const float* emb  = (const float*)d_in[1];
    const float* w_ih = (const float*)d_in[2];
    const float* w_hh = (const float*)d_in[3];
    const float* b_ih = (const float*)d_in[4];
    const float* b_hh = (const float*)d_in[5];
    const float* fc_w = (const float*)d_in[6];
    const float* fc_b = (const float*)d_in[7];
    const float* h0   = (const float*)d_in[8];
    const float* c0   = (const float*)d_in[9];
    float* out = (float*)d_out;

    float* xg = (float*)d_ws;   // B*T*4D f32 = 209,715,200 bytes of scratch

    // Phase 1: 8192 M-tiles / 8 per block * 25 N-tiles = 25600 blocks
    dim3 gridA((Bb * Tt / 16 / 8) * NT);
    lstm_input_gemm<<<gridA, 256, 0, stream>>>(x, emb, w_ih, b_ih, b_hh, xg);

    // Phase 2: 16 persistent scan workgroups, ~200 KB dynamic LDS each
    size_t lds_bytes = (size_t)(Gg * Dh + 3 * ROWS * Dh + ROWS * Gg) * sizeof(float);
    (void)hipFuncSetAttribute((const void*)lstm_scan,
                              hipFuncAttributeMaxDynamicSharedMemorySize,
                              (int)lds_bytes);
    lstm_scan<<<dim3(Bb / ROWS), 256, lds_bytes, stream>>>(
        xg, w_hh, fc_w, fc_b, h0, c0, out);
}